// scaled_dot_product_attention_14766097563709
// MI455X (gfx1250) — compile-verified
//
#include <hip/hip_runtime.h>
#include <stdint.h>

// Flash attention (causal), fp32, CDNA5 gfx1250.
// - Matrix math: V_WMMA_F32_16X16X4_F32 (exact fp32, matches fp32 reference)
// - Data movement: GLOBAL_LOAD_ASYNC_TO_LDS_B128 + double buffering (ASYNCcnt)
// B=2 H=16 S=2048 DK=DV=64.

typedef __attribute__((ext_vector_type(2))) float v2f;
typedef __attribute__((ext_vector_type(8))) float v8f;

#define S_LEN  2048
#define DKV    64
#define BM_WG  128      // query rows per workgroup (8 wave32 x 16 rows)
#define BN     64       // key rows per iteration
#define KSTR   68       // K tile LDS stride (dwords): col-pattern reads conflict-free
#define VSTR   72       // V tile LDS stride (dwords): row-pair reads conflict-free
#define PSTR   68       // P tile LDS stride

__device__ __forceinline__ void async_b128(uint32_t lds_off, const float* gaddr) {
    // DMA 16 bytes per lane: global -> LDS, tracked by ASYNCcnt (no VGPR staging).
    asm volatile("global_load_async_to_lds_b128 %0, %1, off"
                 :: "v"(lds_off), "v"(gaddr) : "memory");
}
__device__ __forceinline__ void wait_async0() {
    asm volatile("s_wait_asynccnt 0x0" ::: "memory");
}

__global__ __launch_bounds__(256) void fa_causal_f32_wmma(
    const float* __restrict__ Q, const float* __restrict__ K,
    const float* __restrict__ V, float* __restrict__ O)
{
    __shared__ __align__(16) float sK[2][BN * KSTR];       // K tile, row-major [key][d]
    __shared__ __align__(16) float sV[2][BN * VSTR];       // V tile, row-major [key][d]
    __shared__ __align__(16) float sP[8 * 16 * PSTR];      // per-wave P tile (C->A relayout)

    const int tid  = threadIdx.x;
    const int lane = tid & 31;
    const int wave = tid >> 5;          // 0..7
    const int nl   = lane & 15;         // M (A) / N (B,C) index
    const int h    = lane >> 4;         // lane half: K-split (A/B), row+8 (C)

    const int blocksPerBH = S_LEN / BM_WG;           // 16
    const int bh    = blockIdx.x / blocksPerBH;      // 0..31
    const int qblk  = blockIdx.x % blocksPerBH;
    const int qbase = qblk * BM_WG;
    const int q0    = qbase + wave * 16;             // this wave's first query row

    const size_t bhoff = (size_t)bh * S_LEN * DKV;
    const float* Qb = Q + bhoff;
    const float* Kb = K + bhoff;
    const float* Vb = V + bhoff;
    float*       Ob = O + bhoff;

    // ---- Q A-fragments, pre-scaled by (1/sqrt(64))*log2(e): scores emerge in
    //      log2-softmax domain, no per-block scaling VALU work. ----
    const float SC = 0.18033688011112042f;
    v2f qf[16];
    {
        const float* qrow = Qb + (size_t)(q0 + nl) * DKV;
        #pragma unroll
        for (int dc = 0; dc < 16; ++dc) {
            v2f q = *reinterpret_cast<const v2f*>(qrow + 4 * dc + 2 * h);
            qf[dc] = q * SC;
        }
    }

    const v8f zero8 = {0.f, 0.f, 0.f, 0.f, 0.f, 0.f, 0.f, 0.f};
    v8f o[4];
    #pragma unroll
    for (int n = 0; n < 4; ++n) o[n] = zero8;

    float mrow[8], lrow[8];             // per-row softmax state (row = q0 + 8h + j)
    #pragma unroll
    for (int j = 0; j < 8; ++j) { mrow[j] = -1e38f; lrow[j] = 0.f; }

    float* sPw = sP + wave * 16 * PSTR;

    const int nkb = 2 * qblk + 2;       // key blocks up to the diagonal (wg-uniform)

    const int lk  = tid >> 2;           // cooperative-copy row  (0..63)
    const int ldo = (tid & 3) * 16;     // cooperative-copy dcol (0/16/32/48)

    // ---- async prefetch of one K/V tile into buffer `buf` ----
    auto issue_tile = [&](int kb, int buf) {
        const float* Kg = Kb + (size_t)(kb * BN + lk) * DKV + ldo;
        const float* Vg = Vb + (size_t)(kb * BN + lk) * DKV + ldo;
        #pragma unroll
        for (int i = 0; i < 4; ++i) {
            async_b128((uint32_t)(uintptr_t)&sK[buf][lk * KSTR + ldo + 4 * i], Kg + 4 * i);
            async_b128((uint32_t)(uintptr_t)&sV[buf][lk * VSTR + ldo + 4 * i], Vg + 4 * i);
        }
    };

    issue_tile(0, 0);
    wait_async0();
    __syncthreads();

    for (int kb = 0; kb < nkb; ++kb) {
        const int cur = kb & 1;

        // prefetch next tile into the other buffer while we compute on this one
        if (kb + 1 < nkb) issue_tile(kb + 1, cur ^ 1);

        if (kb * BN <= q0 + 15) {                    // wave intersects causal region
            const float* sKc = sK[cur];
            const float* sVc = sV[cur];

            // ---- S = Q K^T : 4 tiles of 16x16, K-depth 64 via 16 wmma each ----
            v8f st[4];
            #pragma unroll
            for (int n = 0; n < 4; ++n) {
                v8f sc = zero8;
                #pragma unroll
                for (int dc = 0; dc < 16; ++dc) {
                    v2f b = *reinterpret_cast<const v2f*>(
                        &sKc[(16 * n + nl) * KSTR + 4 * dc + 2 * h]);
                    sc = __builtin_amdgcn_wmma_f32_16x16x4_f32(
                        false, qf[dc], false, b, (short)0, sc, false, false);
                }
                st[n] = sc;
            }

            // ---- causal mask (pure v_cndmask; scale already folded into Q) ----
            #pragma unroll
            for (int n = 0; n < 4; ++n) {
                const int col = kb * BN + 16 * n + nl;
                #pragma unroll
                for (int j = 0; j < 8; ++j) {
                    const int row = q0 + 8 * h + j;
                    st[n][j] = (col <= row) ? st[n][j] : -1e38f;
                }
            }

            // ---- online softmax (row reductions across the 16-lane half) ----
            float corr[8], rowsum[8];
            #pragma unroll
            for (int j = 0; j < 8; ++j) {
                float t = fmaxf(fmaxf(st[0][j], st[1][j]), fmaxf(st[2][j], st[3][j]));
                t = fmaxf(t, __shfl_xor(t, 1, 32));
                t = fmaxf(t, __shfl_xor(t, 2, 32));
                t = fmaxf(t, __shfl_xor(t, 4, 32));
                t = fmaxf(t, __shfl_xor(t, 8, 32));
                const float mnew = fmaxf(mrow[j], t);
                corr[j]   = __builtin_amdgcn_exp2f(mrow[j] - mnew);
                mrow[j]   = mnew;
                rowsum[j] = 0.f;
            }
            #pragma unroll
            for (int n = 0; n < 4; ++n) {
                #pragma unroll
                for (int j = 0; j < 8; ++j) {
                    const float p = __builtin_amdgcn_exp2f(st[n][j] - mrow[j]);
                    st[n][j] = p;
                    rowsum[j] += p;
                }
            }
            #pragma unroll
            for (int j = 0; j < 8; ++j) {
                float r = rowsum[j];
                r += __shfl_xor(r, 1, 32);
                r += __shfl_xor(r, 2, 32);
                r += __shfl_xor(r, 4, 32);
                r += __shfl_xor(r, 8, 32);
                lrow[j] = lrow[j] * corr[j] + r;
                #pragma unroll
                for (int n = 0; n < 4; ++n) o[n][j] *= corr[j];
            }

            // ---- P: C layout -> row-major LDS (per-wave tile, no barrier) ----
            #pragma unroll
            for (int n = 0; n < 4; ++n)
                #pragma unroll
                for (int j = 0; j < 8; ++j)
                    sPw[(8 * h + j) * PSTR + 16 * n + nl] = st[n][j];

            // ---- O += P V : A = P chunks (16x4 over keys), B = V row pairs ----
            #pragma unroll
            for (int kc = 0; kc < 16; ++kc) {
                v2f a = *reinterpret_cast<const v2f*>(
                    &sPw[nl * PSTR + 4 * kc + 2 * h]);
                #pragma unroll
                for (int n = 0; n < 4; ++n) {
                    const float* vp = &sVc[(4 * kc + 2 * h) * VSTR + 16 * n + nl];
                    v2f b = { vp[0], vp[VSTR] };   // rows k, k+1 -> ds_load_2addr_b32
                    o[n] = __builtin_amdgcn_wmma_f32_16x16x4_f32(
                        false, a, false, b, (short)0, o[n], false, false);
                }
            }
        }

        wait_async0();      // our prefetch landed in LDS
        __syncthreads();    // everyone done reading `cur`, everyone's prefetch visible
    }

    // ---- epilogue: normalize by row sums and store ----
    #pragma unroll
    for (int j = 0; j < 8; ++j) {
        const int row = q0 + 8 * h + j;
        const float inv = __builtin_amdgcn_rcpf(lrow[j]);
        float* orow = Ob + (size_t)row * DKV;
        #pragma unroll
        for (int n = 0; n < 4; ++n)
            orow[16 * n + nl] = o[n][j] * inv;
    }
}

extern "C" void kernel_launch(void* const* d_in, const int* in_sizes, int n_in,
                              void* d_out, int out_size, void* d_ws, size_t ws_size,
                              hipStream_t stream) {
    const float* Q = (const float*)d_in[0];
    const float* K = (const float*)d_in[1];
    const float* V = (const float*)d_in[2];
    // d_in[3] is the causal mask; causality is generated analytically.
    float* O = (float*)d_out;

    const int B = 2, H = 16;
    dim3 grid(B * H * (S_LEN / BM_WG));
    fa_causal_f32_wmma<<<grid, 256, 0, stream>>>(Q, K, V, O);
}